// SAGPoolClassifier_8461085573478
// MI455X (gfx1250) — compile-verified
//
#include <hip/hip_runtime.h>
#include <hip/hip_bf16.h>

#define N_NODES 100000
#define N_EDGES 1600000
#define HDIM    128
#define NEGF    (-1e30f)

typedef float v2f __attribute__((ext_vector_type(2)));
typedef float v8f __attribute__((ext_vector_type(8)));

// ---------- helpers ----------
__device__ __forceinline__ unsigned fkey(float f) {
    unsigned u = __float_as_uint(f);
    return (u & 0x80000000u) ? ~u : (u | 0x80000000u);   // monotone-increasing map
}
__device__ __forceinline__ float unfkey(unsigned u) {
    unsigned b = (u & 0x80000000u) ? (u ^ 0x80000000u) : ~u;
    return __uint_as_float(b);
}

// ---------- one-time per-launch init ----------
__global__ void k_init(float* em, float* nm, float* rsum, unsigned* hist, unsigned* ctrl) {
    long long i = (long long)blockIdx.x * blockDim.x + threadIdx.x;
    if (i < N_EDGES) em[i] = 1.0f;
    if (i < N_NODES) nm[i] = 1.0f;
    if (i < 256) { rsum[i] = 0.0f; hist[i] = 0u; }
    if (i < 8) ctrl[i] = 0u;
}

// ---------- degree / dinv ----------
__global__ void k_deg_init(const float* nm, float* deg) {
    int i = blockIdx.x * blockDim.x + threadIdx.x;
    if (i < N_NODES) deg[i] = nm[i];
}
__global__ void k_deg_edges(const float* em, const long long* dst, float* deg) {
    int e = blockIdx.x * blockDim.x + threadIdx.x;
    if (e < N_EDGES) {
        float w = em[e];
        if (w != 0.0f) atomicAdd(&deg[dst[e]], w);
    }
}
__global__ void k_dinv(float* deg) {
    int i = blockIdx.x * blockDim.x + threadIdx.x;
    if (i < N_NODES) {
        float d = deg[i];
        deg[i] = (d > 0.0f) ? rsqrtf(d) : 0.0f;
    }
}

// ---------- aggregation ----------
// agg[i] = nm[i] * dinv[i]^2 * x[i]  (self-loop term preloaded)
__global__ void k_selfloop(const float* __restrict__ x, const float* nm, const float* dinv,
                           float* __restrict__ agg) {
    int i = blockIdx.x, f = threadIdx.x;   // N blocks x 128
    float c = nm[i] * dinv[i] * dinv[i];
    agg[(long long)i * HDIM + f] = c * x[(long long)i * HDIM + f];
}
// agg[dst] += coef * x[src]
__global__ void k_edge_agg(const float* __restrict__ x, const float* em, const float* dinv,
                           const long long* src, const long long* dst,
                           float* __restrict__ agg) {
    int e = blockIdx.x, f = threadIdx.x;   // E blocks x 128
    long long s = src[e], d = dst[e];
    float coef = em[e] * dinv[s] * dinv[d];
    if (coef != 0.0f)
        atomicAdd(&agg[d * HDIM + f], coef * x[s * HDIM + f]);
}

// ---------- GEMM: h = relu(agg @ W + b) * nm   (WMMA f32 16x16x4) ----------
__global__ __launch_bounds__(32) void k_gemm(const float* __restrict__ agg,
                                             const float* __restrict__ W,
                                             const float* __restrict__ b,
                                             const float* __restrict__ nm,
                                             float* __restrict__ h) {
    const int row0 = blockIdx.x * 16;
    const int col0 = blockIdx.y * 16;
    const int lane = threadIdx.x;          // 0..31, wave32
    const int m    = lane & 15;
    const int kh   = lane >> 4;            // 0 or 1 -> K pair select

    float bias = b[col0 + m];
    v8f acc;
#pragma unroll
    for (int r = 0; r < 8; ++r) acc[r] = bias;

    for (int kk = 0; kk < HDIM; kk += 4) {
        int ka = kk + 2 * kh;
        v2f a, bb;
        a[0]  = agg[(long long)(row0 + m) * HDIM + ka];
        a[1]  = agg[(long long)(row0 + m) * HDIM + ka + 1];
        bb[0] = W[(long long)ka * HDIM + col0 + m];
        bb[1] = W[(long long)(ka + 1) * HDIM + col0 + m];
        acc = __builtin_amdgcn_wmma_f32_16x16x4_f32(
            /*neg_a=*/false, a, /*neg_b=*/false, bb,
            /*c_mod=*/(short)0, acc, /*reuse_a=*/false, /*reuse_b=*/false);
    }
#pragma unroll
    for (int r = 0; r < 8; ++r) {
        int row = row0 + r + 8 * kh;       // C layout: VGPR r -> M = r (+8 for hi half)
        float v = acc[r] > 0.0f ? acc[r] : 0.0f;
        h[(long long)row * HDIM + col0 + m] = v * nm[row];
    }
}

// ---------- score GCN (projected to scalar first) ----------
__global__ void k_dot_ws(const float* __restrict__ h, const float* __restrict__ ws, float* p) {
    int i = blockIdx.x, t = threadIdx.x;   // N blocks x 128
    __shared__ float sh[HDIM];
    sh[t] = h[(long long)i * HDIM + t] * ws[t];
    __syncthreads();
    for (int s = 64; s > 0; s >>= 1) {
        if (t < s) sh[t] += sh[t + s];
        __syncthreads();
    }
    if (t == 0) p[i] = sh[0];
}
__global__ void k_score_init(const float* nm, const float* dinv, const float* p, float* sc) {
    int i = blockIdx.x * blockDim.x + threadIdx.x;
    if (i < N_NODES) sc[i] = nm[i] * dinv[i] * dinv[i] * p[i];
}
__global__ void k_score_edges(const float* em, const float* dinv, const float* p,
                              const long long* src, const long long* dst, float* sc) {
    int e = blockIdx.x * blockDim.x + threadIdx.x;
    if (e < N_EDGES) {
        long long s = src[e], d = dst[e];
        float coef = em[e] * dinv[s] * dinv[d];
        if (coef != 0.0f) atomicAdd(&sc[d], coef * p[s]);
    }
}
__global__ void k_score_fin(const float* nm, const float* bs, float* sc) {
    int i = blockIdx.x * blockDim.x + threadIdx.x;
    if (i < N_NODES) sc[i] = (sc[i] + bs[0]) * nm[i];
}

// ---------- exact top-k via 4-round MSB radix select ----------
__global__ void k_hist(const float* sc, const float* nm, unsigned* hist,
                       const unsigned* ctrl, int round) {
    int i = blockIdx.x * blockDim.x + threadIdx.x;
    if (i >= N_NODES) return;
    if (nm[i] <= 0.0f) return;
    unsigned key = fkey(sc[i]);
    if (round > 0 && (key >> (32 - 8 * round)) != ctrl[0]) return;
    atomicAdd(&hist[(key >> (24 - 8 * round)) & 0xFFu], 1u);
}
__global__ void k_pick(unsigned* hist, unsigned* ctrl, int round, unsigned k) {
    if (threadIdx.x == 0) {
        unsigned need = (round == 0) ? k : ctrl[1];
        unsigned cum = 0, bsel = 0;
        for (int bb = 255; bb >= 0; --bb) {
            unsigned c = hist[bb];
            if (cum + c >= need) { bsel = (unsigned)bb; need -= cum; break; }
            cum += c;
        }
        ctrl[0] = (round == 0) ? bsel : ((ctrl[0] << 8) | bsel);
        ctrl[1] = need;
        ctrl[2] = 0u;
    }
    __syncthreads();
    hist[threadIdx.x] = 0u;   // blockDim == 256: clean for next round
}
__global__ void k_select(const float* sc, const float* nm, float* nmNew, unsigned* ctrl) {
    int i = blockIdx.x * blockDim.x + threadIdx.x;
    if (i >= N_NODES) return;
    float out = 0.0f;
    if (nm[i] > 0.0f) {
        unsigned key = fkey(sc[i]);
        unsigned T = ctrl[0];
        if (key > T) out = 1.0f;
        else if (key == T && atomicAdd(&ctrl[2], 1u) < ctrl[1]) out = 1.0f;
    }
    nmNew[i] = out;
}

// ---------- gating + readout ----------
__global__ void k_maxinit(unsigned* maxbuf) {
    if (threadIdx.x < HDIM) maxbuf[threadIdx.x] = fkey(NEGF);
}
__global__ void k_gate(const float* __restrict__ h, const float* sc, const float* nmNew,
                       float* __restrict__ xnew, float* rsum, unsigned* maxbuf, float invk) {
    int f = threadIdx.x;                  // 128 threads
    float lsum = 0.0f, lmax = NEGF;
    for (int i = blockIdx.x; i < N_NODES; i += gridDim.x) {
        float m = nmNew[i];
        float v = h[(long long)i * HDIM + f] * tanhf(sc[i]) * m;
        xnew[(long long)i * HDIM + f] = v;
        if (m > 0.0f) { lsum += v; lmax = fmaxf(lmax, v); }
    }
    atomicAdd(&rsum[f], lsum * invk);
    atomicMax(&maxbuf[f], fkey(lmax));
}
__global__ void k_maxfold(const unsigned* maxbuf, float* rsum) {
    if (threadIdx.x < HDIM) rsum[HDIM + threadIdx.x] += unfkey(maxbuf[threadIdx.x]);
}
__global__ void k_edge_update(const long long* src, const long long* dst,
                              const float* nmNew, float* em) {
    int e = blockIdx.x * blockDim.x + threadIdx.x;
    if (e < N_EDGES) em[e] *= nmNew[src[e]] * nmNew[dst[e]];
}

// ---------- final MLP + log_softmax ----------
__global__ void k_mlp(const float* rsum,
                      const float* M1, const float* bm1,
                      const float* M2, const float* bm2,
                      const float* M3, const float* bm3,
                      float* out) {
    __shared__ float y1[128], y2[64], y3[10], lse;
    int t = threadIdx.x;                  // 256 threads
    if (t < 128) {
        float s = bm1[t];
        for (int i = 0; i < 256; ++i) s += rsum[i] * M1[i * 128 + t];
        y1[t] = s > 0.0f ? s : 0.0f;
    }
    __syncthreads();
    if (t < 64) {
        float s = bm2[t];
        for (int i = 0; i < 128; ++i) s += y1[i] * M2[i * 64 + t];
        y2[t] = s > 0.0f ? s : 0.0f;
    }
    __syncthreads();
    if (t < 10) {
        float s = bm3[t];
        for (int i = 0; i < 64; ++i) s += y2[i] * M3[i * 10 + t];
        y3[t] = s;
    }
    __syncthreads();
    if (t == 0) {
        float mx = y3[0];
        for (int j = 1; j < 10; ++j) mx = fmaxf(mx, y3[j]);
        float se = 0.0f;
        for (int j = 0; j < 10; ++j) se += expf(y3[j] - mx);
        lse = mx + logf(se);
    }
    __syncthreads();
    if (t < 10) out[t] = y3[t] - lse;
}

// ---------- host orchestration ----------
extern "C" void kernel_launch(void* const* d_in, const int* in_sizes, int n_in,
                              void* d_out, int out_size, void* d_ws, size_t ws_size,
                              hipStream_t stream) {
    const float* x  = (const float*)d_in[0];
    const long long* ei = (const long long*)d_in[1];
    const long long* src = ei;
    const long long* dst = ei + N_EDGES;
    const float* W[3]  = { (const float*)d_in[2],  (const float*)d_in[6],  (const float*)d_in[10] };
    const float* b[3]  = { (const float*)d_in[3],  (const float*)d_in[7],  (const float*)d_in[11] };
    const float* ws[3] = { (const float*)d_in[4],  (const float*)d_in[8],  (const float*)d_in[12] };
    const float* bs[3] = { (const float*)d_in[5],  (const float*)d_in[9],  (const float*)d_in[13] };
    const float* M1  = (const float*)d_in[14]; const float* bm1 = (const float*)d_in[15];
    const float* M2  = (const float*)d_in[16]; const float* bm2 = (const float*)d_in[17];
    const float* M3  = (const float*)d_in[18]; const float* bm3 = (const float*)d_in[19];
    float* out = (float*)d_out;

    // workspace carve-up
    char* p = (char*)d_ws;
    float* B0   = (float*)p; p += (size_t)N_NODES * HDIM * sizeof(float);
    float* B1   = (float*)p; p += (size_t)N_NODES * HDIM * sizeof(float);
    float* EM   = (float*)p; p += (size_t)N_EDGES * sizeof(float);
    float* NMA  = (float*)p; p += (size_t)N_NODES * sizeof(float);
    float* NMB  = (float*)p; p += (size_t)N_NODES * sizeof(float);
    float* DINV = (float*)p; p += (size_t)N_NODES * sizeof(float);
    float* PRJ  = (float*)p; p += (size_t)N_NODES * sizeof(float);
    float* SC   = (float*)p; p += (size_t)N_NODES * sizeof(float);
    float* RSUM = (float*)p; p += 256 * sizeof(float);
    unsigned* MAXB = (unsigned*)p; p += 128 * sizeof(unsigned);
    unsigned* HIST = (unsigned*)p; p += 256 * sizeof(unsigned);
    unsigned* CTRL = (unsigned*)p; p += 32 * sizeof(unsigned);

    const int TB = 256;
    const int gN = (N_NODES + TB - 1) / TB;
    const int gE = (N_EDGES + TB - 1) / TB;

    k_init<<<gE, TB, 0, stream>>>(EM, NMA, RSUM, HIST, CTRL);

    const unsigned kk[3] = { 50000u, 25000u, 12500u };
    const float*  xin[3]  = { x,  B0, B1 };
    float*        aggb[3] = { B0, B1, B0 };
    float*        hb[3]   = { B1, B0, B1 };
    float*        xnew[3] = { B0, B1, B0 };
    float*        nmc[3]  = { NMA, NMB, NMA };
    float*        nmn[3]  = { NMB, NMA, NMB };

    for (int L = 0; L < 3; ++L) {
        // degree + dinv
        k_deg_init<<<gN, TB, 0, stream>>>(nmc[L], DINV);
        k_deg_edges<<<gE, TB, 0, stream>>>(EM, dst, DINV);
        k_dinv<<<gN, TB, 0, stream>>>(DINV);
        // aggregate
        k_selfloop<<<N_NODES, HDIM, 0, stream>>>(xin[L], nmc[L], DINV, aggb[L]);
        k_edge_agg<<<N_EDGES, HDIM, 0, stream>>>(xin[L], EM, DINV, src, dst, aggb[L]);
        // h = relu(agg @ W + b) * nm   (WMMA)
        dim3 gg(N_NODES / 16, HDIM / 16);
        k_gemm<<<gg, 32, 0, stream>>>(aggb[L], W[L], b[L], nmc[L], hb[L]);
        // scalar score GCN (projection trick)
        k_dot_ws<<<N_NODES, HDIM, 0, stream>>>(hb[L], ws[L], PRJ);
        k_score_init<<<gN, TB, 0, stream>>>(nmc[L], DINV, PRJ, SC);
        k_score_edges<<<gE, TB, 0, stream>>>(EM, DINV, PRJ, src, dst, SC);
        k_score_fin<<<gN, TB, 0, stream>>>(nmc[L], bs[L], SC);
        // exact top-k (radix select, MSB-first, 4 bytes)
        for (int r = 0; r < 4; ++r) {
            k_hist<<<gN, TB, 0, stream>>>(SC, nmc[L], HIST, CTRL, r);
            k_pick<<<1, 256, 0, stream>>>(HIST, CTRL, r, kk[L]);
        }
        k_select<<<gN, TB, 0, stream>>>(SC, nmc[L], nmn[L], CTRL);
        // gating + readout
        k_maxinit<<<1, 128, 0, stream>>>(MAXB);
        k_gate<<<1024, HDIM, 0, stream>>>(hb[L], SC, nmn[L], xnew[L], RSUM, MAXB,
                                          1.0f / (float)kk[L]);
        k_maxfold<<<1, 128, 0, stream>>>(MAXB, RSUM);
        // filter edges for next level
        k_edge_update<<<gE, TB, 0, stream>>>(src, dst, nmn[L], EM);
    }

    k_mlp<<<1, 256, 0, stream>>>(RSUM, M1, bm1, M2, bm2, M3, bm3, out);
    (void)in_sizes; (void)n_in; (void)out_size; (void)ws_size;
}